// RetinaStaticExportWrapper_10857677324962
// MI455X (gfx1250) — compile-verified
//
#include <hip/hip_runtime.h>
#include <math.h>

#define CONF_THR 0.6f
#define NMS_THR  0.4f
#define TOPK     5000
#define NPAD     5024            // padded slots, multiple of 32
#define WRDS     (NPAD / 32)     // 157 bitmask words per row
#define CHUNK    2048            // elements per compaction block
#define MAXNB    4096

typedef float v2f __attribute__((ext_vector_type(2)));
typedef float v8f __attribute__((ext_vector_type(8)));

// ---------------------------------------------------------------- init
__global__ __launch_bounds__(256) void init_kernel(int* sel_idx, float* zbase, int nz) {
    int i = blockIdx.x * blockDim.x + threadIdx.x;
    if (i < NPAD) sel_idx[i] = -1;
    for (int k = i; k < nz; k += gridDim.x * blockDim.x) zbase[k] = 0.f;
}

// ------------------------------------------------- compaction: count
__global__ __launch_bounds__(256) void count_kernel(const float* __restrict__ conf,
                                                    int* __restrict__ blkCnt, int n) {
    __shared__ int s_sum[8];
    const int tid = threadIdx.x, lane = tid & 31, warp = tid >> 5;
    const int base = blockIdx.x * CHUNK;
    int c = 0;
    #pragma unroll
    for (int r = 0; r < CHUNK / 256; ++r) {
        int i = base + r * 256 + tid;
        if (r + 1 < CHUNK / 256) {
            int ip = base + (r + 1) * 256 + tid;
            if (ip < n) __builtin_prefetch(conf + (size_t)ip * 2 + 1, 0, 1);
        }
        c += (i < n && conf[(size_t)i * 2 + 1] > CONF_THR) ? 1 : 0;
    }
    for (int off = 16; off > 0; off >>= 1) c += __shfl_down(c, off, 32);
    if (lane == 0) s_sum[warp] = c;
    __syncthreads();
    if (tid == 0) {
        int t = 0;
        for (int w = 0; w < 8; ++w) t += s_sum[w];
        blkCnt[blockIdx.x] = t;
    }
}

// -------------------------------------------- compaction: block scan
__global__ void scan_kernel(const int* __restrict__ blkCnt, int* __restrict__ blkOff, int nb) {
    if (threadIdx.x == 0 && blockIdx.x == 0) {
        int run = 0;
        for (int b = 0; b < nb; ++b) { blkOff[b] = run; run += blkCnt[b]; }
    }
}

// ------------------------------------------ compaction: ordered select
__global__ __launch_bounds__(256) void select_kernel(const float* __restrict__ conf,
                                                     const int* __restrict__ blkOff,
                                                     int* __restrict__ sel_idx, int n) {
    __shared__ int s_wcnt[8];
    __shared__ int s_running;
    const int tid = threadIdx.x, lane = tid & 31, warp = tid >> 5;
    if (tid == 0) s_running = blkOff[blockIdx.x];
    __syncthreads();
    const int base = blockIdx.x * CHUNK;
    #pragma unroll
    for (int r = 0; r < CHUNK / 256; ++r) {
        int i = base + r * 256 + tid;
        bool pred = (i < n) && (conf[(size_t)i * 2 + 1] > CONF_THR);
        unsigned bal = (unsigned)__ballot(pred);          // wave32: low 32 bits
        if (lane == 0) s_wcnt[warp] = __popc(bal);
        __syncthreads();
        int wpre = 0, rtot = 0;
        for (int w = 0; w < 8; ++w) { if (w < warp) wpre += s_wcnt[w]; rtot += s_wcnt[w]; }
        int lpre = __popc(bal & ((1u << lane) - 1u));
        int pos = s_running + wpre + lpre;
        if (pred && pos < TOPK) sel_idx[pos] = i;
        __syncthreads();
        if (tid == 0) s_running += rtot;
        __syncthreads();
    }
}

// --------------------------------------------------- lazy decode (5000)
__global__ __launch_bounds__(256) void decode_kernel(const float* __restrict__ loc,
                                                     const float* __restrict__ conf,
                                                     const float* __restrict__ landms,
                                                     const float* __restrict__ priors,
                                                     const int* __restrict__ psx,
                                                     const int* __restrict__ psy,
                                                     const int* __restrict__ sel_idx,
                                                     float* __restrict__ boxes_sel,
                                                     float* __restrict__ landms_sel,
                                                     float* __restrict__ conf_sel,
                                                     float* __restrict__ area, int P) {
    int s = blockIdx.x * blockDim.x + threadIdx.x;
    if (s >= TOPK) return;
    int i = sel_idx[s];
    if (i < 0) return;                                   // slot invalid: stays zero
    const float sx = (float)psx[0], sy = (float)psy[0];
    const int pi = i % P;
    const float* pr = priors + (size_t)pi * 4;
    const float* lc = loc + (size_t)i * 4;
    const float pcx = pr[0], pcy = pr[1], pw = pr[2], ph = pr[3];
    float cx = pcx + lc[0] * 0.1f * pw;
    float cy = pcy + lc[1] * 0.1f * ph;
    float w  = pw * expf(lc[2] * 0.2f);
    float h  = ph * expf(lc[3] * 0.2f);
    float x1 = (cx - 0.5f * w) * sx, y1 = (cy - 0.5f * h) * sy;
    float x2 = (cx + 0.5f * w) * sx, y2 = (cy + 0.5f * h) * sy;
    boxes_sel[s * 4 + 0] = x1; boxes_sel[s * 4 + 1] = y1;
    boxes_sel[s * 4 + 2] = x2; boxes_sel[s * 4 + 3] = y2;
    area[s] = fmaxf(x2 - x1, 0.f) * fmaxf(y2 - y1, 0.f);
    const float* lm = landms + (size_t)i * 10;
    #pragma unroll
    for (int k = 0; k < 5; ++k) {
        landms_sel[s * 10 + 2 * k]     = (pcx + pw * lm[2 * k]     * 0.1f) * sx;
        landms_sel[s * 10 + 2 * k + 1] = (pcy + ph * lm[2 * k + 1] * 0.1f) * sy;
    }
    conf_sel[s * 2 + 0] = conf[(size_t)i * 2 + 0];
    conf_sel[s * 2 + 1] = conf[(size_t)i * 2 + 1];
}

// ---------------- IoU suppression bitmask: one wave per 16x32 tile ----
// area_i + area_j outer sum via V_WMMA_F32_16X16X4_F32:
//   A (16x4): col0 = area_i, col1 = 1      B (4x16): row0 = 1, row1 = area_j
// bj tile staged through LDS with GLOBAL_LOAD_ASYNC_TO_LDS_B128 (ASYNCcnt).
__global__ __launch_bounds__(32) void iou_mask_kernel(const float* __restrict__ boxes,
                                                      const float* __restrict__ area,
                                                      unsigned* __restrict__ supp) {
    const int ti = blockIdx.x;           // row tile: rows [ti*16, ti*16+15]
    const int wc = blockIdx.y;           // word col: cols [wc*32, wc*32+31]
    const int lane = threadIdx.x;
    const int rowBase = ti * 16;
    const int colBase = wc * 32;

    __shared__ float bi[16][4];
    __shared__ float bj[32][4];

    // async per-lane 16B copy: boxes[colBase+lane] -> bj[lane]  (512B tile)
    {
        unsigned long long ga = (unsigned long long)(uintptr_t)(boxes + (size_t)(colBase + lane) * 4);
        unsigned lo = (unsigned)(uintptr_t)&bj[lane][0];
        asm volatile("global_load_async_to_lds_b128 %0, %1, off"
                     :: "v"(lo), "v"(ga) : "memory");
    }
    if (lane < 16) {
        const float* p = boxes + (size_t)(rowBase + lane) * 4;
        bi[lane][0] = p[0]; bi[lane][1] = p[1]; bi[lane][2] = p[2]; bi[lane][3] = p[3];
    }

    // A layout: VGPR0 = K0(lanes 0-15)/K2(16-31), VGPR1 = K1/K3
    v2f A;
    A.x = (lane < 16) ? area[rowBase + lane] : 0.f;   // K=0 col: area_i
    A.y = (lane < 16) ? 1.f : 0.f;                    // K=1 col: ones
    // B layout mirrors: VGPR0 = row K0/K2, VGPR1 = row K1/K3
    v2f B0, B1;
    B0.x = (lane < 16) ? 1.f : 0.f;
    B0.y = (lane < 16) ? area[colBase + lane] : 0.f;
    B1.x = (lane < 16) ? 1.f : 0.f;
    B1.y = (lane < 16) ? area[colBase + 16 + lane] : 0.f;

    v8f C0 = {}; v8f C1 = {};
    C0 = __builtin_amdgcn_wmma_f32_16x16x4_f32(false, A, false, B0, (short)0, C0, false, false);
    C1 = __builtin_amdgcn_wmma_f32_16x16x4_f32(false, A, false, B1, (short)0, C1, false, false);

    asm volatile("s_wait_asynccnt 0" ::: "memory");    // bj tile landed in LDS
    __syncthreads();                                   // bi stores visible (DScnt)

    const int nloc = lane & 15;
    const int mhi  = (lane >> 4) * 8;       // C layout: lanes 16-31 hold rows M+8
    const float jx1a = bj[nloc][0],      jy1a = bj[nloc][1],
                jx2a = bj[nloc][2],      jy2a = bj[nloc][3];
    const float jx1b = bj[16 + nloc][0], jy1b = bj[16 + nloc][1],
                jx2b = bj[16 + nloc][2], jy2b = bj[16 + nloc][3];
    const int col0 = colBase + nloc;
    const int col1 = colBase + 16 + nloc;

    #pragma unroll
    for (int v = 0; v < 8; ++v) {
        const int M = v + mhi;
        const int row = rowBase + M;
        const float ix1 = bi[M][0], iy1 = bi[M][1], ix2 = bi[M][2], iy2 = bi[M][3];

        float iw0 = fminf(ix2, jx2a) - fmaxf(ix1, jx1a);
        float ih0 = fminf(iy2, jy2a) - fmaxf(iy1, jy1a);
        float in0 = fmaxf(iw0, 0.f) * fmaxf(ih0, 0.f);
        float un0 = fmaxf(C0[v] - in0, 1e-12f);           // area_i+area_j from WMMA
        // inter/union > thr  <=>  inter > thr*union   (union > 0) — no FP divide
        bool bit0 = (in0 > NMS_THR * un0) && (col0 > row);
        unsigned bal0 = (unsigned)__ballot(bit0);

        float iw1 = fminf(ix2, jx2b) - fmaxf(ix1, jx1b);
        float ih1 = fminf(iy2, jy2b) - fmaxf(iy1, jy1b);
        float in1 = fmaxf(iw1, 0.f) * fmaxf(ih1, 0.f);
        float un1 = fmaxf(C1[v] - in1, 1e-12f);
        bool bit1 = (in1 > NMS_THR * un1) && (col1 > row);
        unsigned bal1 = (unsigned)__ballot(bit1);

        // ballot low16 = row v (cols 0-15 of half), high16 = row v+8
        if (lane == 0) {
            supp[(size_t)(rowBase + v)     * WRDS + wc] = (bal0 & 0xffffu) | (bal1 << 16);
            supp[(size_t)(rowBase + v + 8) * WRDS + wc] = (bal0 >> 16) | (bal1 & 0xffff0000u);
        }
    }
}

// -------------------- serial greedy NMS scan, single wave32, LDS keep --
__global__ __launch_bounds__(32) void nms_scan_kernel(const int* __restrict__ sel_idx,
                                                      const unsigned* __restrict__ supp,
                                                      unsigned* __restrict__ keep_out) {
    __shared__ unsigned kw[WRDS];
    const int lane = threadIdx.x;
    for (int w = lane; w < WRDS; w += 32) {
        unsigned m = 0;
        for (int b = 0; b < 32; ++b) {
            int s = w * 32 + b;
            if (s < TOPK && sel_idx[s] >= 0) m |= (1u << b);
        }
        kw[w] = m;
    }
    __syncthreads();
    for (int i = 0; i < TOPK; ++i) {
        // hide L2 latency of upcoming suppression rows (628B each)
        if (i + 8 < TOPK && lane < 5)
            __builtin_prefetch(supp + (size_t)(i + 8) * WRDS + lane * 32, 0, 1);
        unsigned word = kw[i >> 5];                      // uniform read
        if ((word >> (i & 31)) & 1u) {
            const unsigned* row = supp + (size_t)i * WRDS;
            for (int w = lane; w < WRDS; w += 32) kw[w] &= ~row[w];
        }
        __syncthreads();
    }
    for (int w = lane; w < WRDS; w += 32) keep_out[w] = kw[w];
}

// ------------------------------------------------------------ finalize
__global__ __launch_bounds__(256) void finalize_kernel(const float* __restrict__ conf_sel,
                                                       const float* __restrict__ landms_sel,
                                                       const float* __restrict__ boxes_sel,
                                                       const unsigned* __restrict__ keepw,
                                                       float* __restrict__ out) {
    int s = blockIdx.x * blockDim.x + threadIdx.x;
    if (s >= TOPK) return;
    float m = ((keepw[s >> 5] >> (s & 31)) & 1u) ? 1.f : 0.f;
    out[s * 2 + 0] = conf_sel[s * 2 + 0] * m;            // cs  @ 0
    out[s * 2 + 1] = conf_sel[s * 2 + 1] * m;
    float* p10 = out + TOPK * 2;                         // p10 @ 10000
    #pragma unroll
    for (int k = 0; k < 10; ++k) p10[s * 10 + k] = landms_sel[s * 10 + k] * m;
    float* b = out + TOPK * 12;                          // b   @ 60000
    #pragma unroll
    for (int k = 0; k < 4; ++k) b[s * 4 + k] = boxes_sel[s * 4 + k] * m;
    out[TOPK * 16 + s] = m;                              // keep @ 80000
}

// ----------------------------------------------------------------------
extern "C" void kernel_launch(void* const* d_in, const int* in_sizes, int n_in,
                              void* d_out, int out_size, void* d_ws, size_t ws_size,
                              hipStream_t stream) {
    const float* loc    = (const float*)d_in[0];
    const float* conf   = (const float*)d_in[1];
    const float* landms = (const float*)d_in[2];
    const float* priors = (const float*)d_in[3];
    const int*   psx    = (const int*)d_in[4];
    const int*   psy    = (const int*)d_in[5];
    const int n = in_sizes[1] / 2;       // B*P
    const int P = in_sizes[3] / 4;

    // workspace layout (4-byte units)
    int* ws            = (int*)d_ws;
    int* blkCnt        = ws;
    int* blkOff        = ws + MAXNB;
    int* sel_idx       = ws + 2 * MAXNB;
    float* conf_sel    = (float*)(ws + 2 * MAXNB + NPAD);
    float* landms_sel  = conf_sel + (size_t)NPAD * 2;
    float* boxes_sel   = landms_sel + (size_t)NPAD * 10;
    float* area        = boxes_sel + (size_t)NPAD * 4;
    unsigned* keepw    = (unsigned*)(area + NPAD);
    unsigned* supp     = keepw + 160;

    const int nb = (n + CHUNK - 1) / CHUNK;

    init_kernel<<<336, 256, 0, stream>>>(sel_idx, conf_sel, NPAD * 17 + 160);
    count_kernel<<<nb, 256, 0, stream>>>(conf, blkCnt, n);
    scan_kernel<<<1, 32, 0, stream>>>(blkCnt, blkOff, nb);
    select_kernel<<<nb, 256, 0, stream>>>(conf, blkOff, sel_idx, n);
    decode_kernel<<<(TOPK + 255) / 256, 256, 0, stream>>>(
        loc, conf, landms, priors, psx, psy, sel_idx,
        boxes_sel, landms_sel, conf_sel, area, P);
    dim3 g(NPAD / 16, WRDS);
    iou_mask_kernel<<<g, 32, 0, stream>>>(boxes_sel, area, supp);
    nms_scan_kernel<<<1, 32, 0, stream>>>(sel_idx, supp, keepw);
    finalize_kernel<<<(TOPK + 255) / 256, 256, 0, stream>>>(
        conf_sel, landms_sel, boxes_sel, keepw, (float*)d_out);
}